// GINLayer_17411797418332
// MI455X (gfx1250) — compile-verified
//
#include <hip/hip_runtime.h>

typedef __attribute__((ext_vector_type(2))) float v2f;
typedef __attribute__((ext_vector_type(8))) float v8f;

#define IN_CH 64
#define HID   128

// ---------------------------------------------------------------------------
// Kernel 1: agg = x  (GIN eps=0: h_i = x_i + sum_j x_j, so seed agg with x)
// ---------------------------------------------------------------------------
__global__ __launch_bounds__(256) void gin_init_kernel(const float4* __restrict__ x,
                                                       float4* __restrict__ agg,
                                                       int n4) {
  int i = blockIdx.x * 256 + threadIdx.x;
  if (i < n4) agg[i] = x[i];
}

// ---------------------------------------------------------------------------
// Kernel 2: scatter-add  agg[dst] += x[src]
// One thread per (edge, channel). 64 consecutive threads share an edge, so the
// index loads are broadcast and x/agg traffic is coalesced. x and agg are both
// L2-resident (12.8 MB each << 192 MB L2), so this runs at L2 atomic rate.
// ---------------------------------------------------------------------------
__global__ __launch_bounds__(256) void gin_scatter_kernel(const float* __restrict__ x,
                                                          const int* __restrict__ src,
                                                          const int* __restrict__ dst,
                                                          float* __restrict__ agg,
                                                          int n_edges) {
  long long tid = (long long)blockIdx.x * 256 + threadIdx.x;
  int edge = (int)(tid >> 6);
  int ch   = (int)(tid & 63);
  if (edge < n_edges) {
    int s = src[edge];
    int d = dst[edge];
    atomicAdd(agg + (long long)d * IN_CH + ch, x[(long long)s * IN_CH + ch]);
  }
}

// ---------------------------------------------------------------------------
// Kernel 3: fused MLP:  out = tanh(relu(h@W1^T + b1)@W2^T + b2)
// Block = 256 threads = 8 wave32, covers 16 nodes.
// Wave w computes output-channel tile n0 = 16*w.
// GEMM1: 16 x v_wmma_f32_16x16x4_f32 (K=64), GEMM2: 32 (K=128).
// WMMA f32 layouts (ISA 7.12.2):
//   A (16x4): lane l<16 -> M=l, VGPR j -> K=j; lanes 16..31 -> K=j+2.
//   B (4x16): lane l<16 -> N=l, VGPR j -> K=j; lanes 16..31 -> K=j+2.
//   C/D (16x16): lane l -> N=l%16, VGPR v -> M = v + 8*(l/16).
// ---------------------------------------------------------------------------
__global__ __launch_bounds__(256) void gin_mlp_kernel(const float* __restrict__ h,
                                                      const float* __restrict__ w1,
                                                      const float* __restrict__ b1,
                                                      const float* __restrict__ w2,
                                                      const float* __restrict__ b2,
                                                      float* __restrict__ out,
                                                      int n_nodes) {
  __shared__ float h1[16 * 130];  // 16 nodes x 128 hidden, stride 130 (even -> b64 aligned)

  const int lane = threadIdx.x & 31;
  const int wave = threadIdx.x >> 5;   // 0..7
  const int half = lane >> 4;          // 0: K+0/K+1,  1: K+2/K+3
  const int l16  = lane & 15;
  const int n0   = wave * 16;
  const long long node_base = (long long)blockIdx.x * 16;

  long long row = node_base + l16;
  if (row >= n_nodes) row = n_nodes - 1;   // clamp loads; stores are guarded

  // ---- GEMM1: C[16 nodes x 16 hid] = H_tile @ W1_tile^T ----
  const float* arow = h  + row * IN_CH + 2 * half;                       // H[row][k + 2*half]
  const float* brow = w1 + (long long)(n0 + l16) * IN_CH + 2 * half;     // W1[n][k + 2*half]

  v8f c = {0.f, 0.f, 0.f, 0.f, 0.f, 0.f, 0.f, 0.f};
#pragma unroll
  for (int k = 0; k < IN_CH; k += 4) {
    v2f a = *(const v2f*)(arow + k);   // global_load_b64
    v2f b = *(const v2f*)(brow + k);   // global_load_b64 (contiguous in row-major W1)
    c = __builtin_amdgcn_wmma_f32_16x16x4_f32(false, a, false, b, (short)0, c,
                                              false, false);
  }

  const float bias1 = b1[n0 + l16];
#pragma unroll
  for (int v = 0; v < 8; ++v) {
    int m = v + 8 * half;
    h1[m * 130 + n0 + l16] = fmaxf(c[v] + bias1, 0.0f);   // bias + ReLU -> LDS
  }
  __syncthreads();

  // ---- GEMM2: C2[16 nodes x 16 out] = H1 @ W2_tile^T ----
  const float* a2   = &h1[l16 * 130 + 2 * half];                         // H1[m][k + 2*half]
  const float* b2r  = w2 + (long long)(n0 + l16) * HID + 2 * half;       // W2[n][k + 2*half]

  v8f c2 = {0.f, 0.f, 0.f, 0.f, 0.f, 0.f, 0.f, 0.f};
#pragma unroll
  for (int k = 0; k < HID; k += 4) {
    v2f a = *(const v2f*)(a2 + k);     // ds_load_b64
    v2f b = *(const v2f*)(b2r + k);    // global_load_b64, L2-hot (W2 = 64 KB)
    c2 = __builtin_amdgcn_wmma_f32_16x16x4_f32(false, a, false, b, (short)0, c2,
                                               false, false);
  }

  const float bias2 = b2[n0 + l16];
#pragma unroll
  for (int v = 0; v < 8; ++v) {
    long long node = node_base + v + 8 * half;
    if (node < n_nodes)
      out[node * HID + n0 + l16] = tanhf(c2[v] + bias2);
  }
}

// ---------------------------------------------------------------------------
extern "C" void kernel_launch(void* const* d_in, const int* in_sizes, int n_in,
                              void* d_out, int out_size, void* d_ws, size_t ws_size,
                              hipStream_t stream) {
  const float* x  = (const float*)d_in[0];
  const int*   ei = (const int*)d_in[1];   // [2, n_edges] flat: src then dst
  const float* w1 = (const float*)d_in[2];
  const float* b1 = (const float*)d_in[3];
  const float* w2 = (const float*)d_in[4];
  const float* b2 = (const float*)d_in[5];
  float* out = (float*)d_out;

  const int n_nodes = in_sizes[0] / IN_CH;
  const int n_edges = in_sizes[1] / 2;

  float* agg = (float*)d_ws;  // n_nodes * IN_CH floats of scratch

  // 1) agg = x
  const int n4 = (n_nodes * IN_CH) / 4;
  gin_init_kernel<<<(n4 + 255) / 256, 256, 0, stream>>>(
      (const float4*)x, (float4*)agg, n4);

  // 2) agg[dst] += x[src]   (one lane per edge-channel)
  const long long work = (long long)n_edges * IN_CH;
  const int sblocks = (int)((work + 255) / 256);
  gin_scatter_kernel<<<sblocks, 256, 0, stream>>>(x, ei, ei + n_edges, agg, n_edges);

  // 3) fused WMMA MLP
  const int mblocks = (n_nodes + 15) / 16;
  gin_mlp_kernel<<<mblocks, 256, 0, stream>>>(agg, w1, b1, w2, b2, out, n_nodes);
}